// GolbalPositionIntersectConcatAttention_40484361732520
// MI455X (gfx1250) — compile-verified
//
#include <hip/hip_runtime.h>
#include <hip/hip_bf16.h>

// Problem constants (match reference)
constexpr int kB = 2048;
constexpr int kN = 512;
constexpr int kD = 64;
constexpr int kK = 4;              // NUM_ATTN
constexpr float kSlope = 0.01f;
constexpr int kRS = 68;            // padded LDS row stride (floats) -> bank-conflict-free WMMA A reads
constexpr int kThreads = 256;      // 8 waves (wave32)

typedef float v2f __attribute__((ext_vector_type(2)));
typedef float v8f __attribute__((ext_vector_type(8)));
typedef unsigned int v4u __attribute__((ext_vector_type(4)));
typedef int v4i __attribute__((ext_vector_type(4)));
typedef int v8i __attribute__((ext_vector_type(8)));

#if defined(__HIP_DEVICE_COMPILE__) && __has_builtin(__builtin_amdgcn_tensor_load_to_lds)
#define USE_TDM 1
#else
#define USE_TDM 0
#endif

// ---------------------------------------------------------------------------
// TDM: one tensor_load_to_lds moves a full 512x64 f32 tile Global->LDS with
// hardware padding of 4 DWORDs every 64 DWORDs (row stride 68 floats).
// D# bit layout per CDNA5 ISA ch.8 (group0 128b, group1 256b).
// ---------------------------------------------------------------------------
#if USE_TDM
__device__ __forceinline__ void tdm_load_tile(unsigned lds_off, const float* gsrc) {
  unsigned long long ga = (unsigned long long)(size_t)gsrc;
  v4u g0;
  g0[0] = 1u;                                               // count=1 (valid user D#)
  g0[1] = lds_off;                                          // lds_addr (bytes)
  g0[2] = (unsigned)(ga & 0xFFFFFFFFu);                     // global_addr[31:0]
  g0[3] = (unsigned)((ga >> 32) & 0x1FFFFFFu) | (2u << 30); // global_addr[56:32], type=2
  v8i g1;
  g1[0] = (int)((2u << 16)      // data_size = 4 bytes
              | (1u << 20)      // pad_enable
              | (5u << 22)      // pad_interval: every 64 DWORDs (one 64-float row)
              | (3u << 25));    // pad_amount: 4 DWORDs -> 68-float row stride
  g1[1] = (int)(64u << 16);     // tensor_dim0 = 64   (bits 79:48, low half)
  g1[2] = (int)(512u << 16);    // tensor_dim1 = 512  (bits 111:80, low half)
  g1[3] = (int)(64u << 16);     // tile_dim0 = 64     (bits 127:112)
  g1[4] = (int)512;             // tile_dim1 = 512    (bits 143:128); tile_dim2 = 0
  g1[5] = 64;                   // tensor_dim0_stride = 64 (bits 207:160, low 32)
  g1[6] = 0;                    // dim0_stride hi, dim1_stride lo
  g1[7] = 0;                    // dim1_stride hi
  v4i z4 = {0, 0, 0, 0};
#if defined(__clang_major__) && __clang_major__ >= 23
  v8i z8 = {0, 0, 0, 0, 0, 0, 0, 0};
  __builtin_amdgcn_tensor_load_to_lds(g0, g1, z4, z4, z8, 0);
#else
  __builtin_amdgcn_tensor_load_to_lds(g0, g1, z4, z4, 0);
#endif
}
#endif

// Fallback: cooperative, fully coalesced global->LDS tile load.
__device__ __forceinline__ void load_tile(float* __restrict__ tile,
                                          const float* __restrict__ src, int t) {
  const float4* s4 = (const float4*)src;
  #pragma unroll
  for (int i = 0; i < (kN * kD / 4) / kThreads; ++i) {
    int e4 = t + i * kThreads;
    float4 v = s4[e4];
    int row = e4 >> 4;
    int c4  = (e4 & 15) << 2;
    *(float4*)(&tile[row * kRS + c4]) = v;
  }
}

// One GEMM phase: acc[j] += tile(4 tiles of 16x64) x Wpad(64x16, zero-padded)
// via chained V_WMMA_F32_16X16X4_F32 over the D=64 reduction dim.
// B operand read from LDS: lane m, rows (d0+2h, d0+2h+1) -> conflict-free.
__device__ __forceinline__ void gemm_phase(const float* __restrict__ tile,
                                           const float* __restrict__ wlds, // 64x16
                                           int wave, int m, int h, v8f acc[4]) {
  #pragma unroll
  for (int d0 = 0; d0 < kD; d0 += 4) {
    const float* bp = wlds + (d0 + 2 * h) * 16 + m;
    v2f bm; bm.x = bp[0]; bm.y = bp[16];
    #pragma unroll
    for (int j = 0; j < 4; ++j) {
      const float* ap = tile + (((wave * 4 + j) * 16 + m) * kRS) + d0 + 2 * h;
      v2f am; am.x = ap[0]; am.y = ap[1];
      acc[j] = __builtin_amdgcn_wmma_f32_16x16x4_f32(
          false, am, false, bm, (short)0, acc[j], false, false);
    }
  }
}

__global__ __launch_bounds__(kThreads)
void gpica_attn_kernel(const float* __restrict__ key_embeds,
                       const float* __restrict__ key_spa,
                       const float* __restrict__ qg_spa,
                       const float* __restrict__ q_embed,
                       const float* __restrict__ av,       // (4D, 4) row-major
                       float* __restrict__ out) {
  extern __shared__ float lds[];
  float* tile = lds;                      // 512*68 floats (139 KB)
  float* expv = tile + kN * kRS;          // 512*4
  float* red  = expv + kN * kK;           // 256
  float* warr = red + kThreads;           // 512
  float* qb   = warr + kN;                // 4
  float* wpad = qb + kK;                  // 2 * 64*16 (Wk then Ws, zero-padded)

  const int b    = blockIdx.x;
  const int t    = threadIdx.x;
  const int lane = t & 31;
  const int wave = t >> 5;                // 0..7
  const int m    = lane & 15;
  const int h    = lane >> 4;             // 0/1

  const float* keyb = key_embeds + (size_t)b * kN * kD;
  const float* spab = key_spa    + (size_t)b * kN * kD;

  // ---- phase A: stage spa tile ----
#if USE_TDM
  if (wave == 0) {
    tdm_load_tile((unsigned)(size_t)tile, spab);
    __builtin_amdgcn_s_wait_tensorcnt(0);
  }
#else
  load_tile(tile, spab, t);
#endif

  // ---- stage zero-padded Wk / Ws into LDS (kills divergent B loads) ----
  #pragma unroll
  for (int i = t; i < 2 * kD * 16; i += kThreads) {
    int mat = i >> 10;                    // 0 = Wk, 1 = Ws
    int rc  = i & 1023;
    int row = rc >> 4, col = rc & 15;
    wpad[i] = (col < kK) ? av[((mat ? 2 * kD : kD) + row) * kK + col] : 0.f;
  }

  // ---- per-batch query bias qb[k] = q_embed.Wq + qg_spa.Wg (tiny) ----
  if (t < kK) {
    float s = 0.f;
    #pragma unroll 4
    for (int d = 0; d < kD; ++d) {
      s += q_embed[b * kD + d] * av[(0 * kD + d) * kK + t];   // Wq
      s += qg_spa[b * kD + d]  * av[(3 * kD + d) * kK + t];   // Wg
    }
    qb[t] = s;
  }

  // prefetch the key tile while spa phase computes
  #pragma unroll
  for (int i = 0; i < 4; ++i)
    __builtin_prefetch(keyb + (size_t)(t + i * kThreads) * 32, 0, 1);
  __syncthreads();

  v8f acc[4] = {v8f{}, v8f{}, v8f{}, v8f{}};
  gemm_phase(tile, wpad + 1024, wave, m, h, acc);   // spa x Ws
  __syncthreads();                                  // all waves done reading spa

  // ---- phase B: stage key tile into the same buffer, accumulate key x Wk ----
#if USE_TDM
  if (wave == 0) {
    tdm_load_tile((unsigned)(size_t)tile, keyb);
    __builtin_amdgcn_s_wait_tensorcnt(0);
  }
#else
  load_tile(tile, keyb, t);
#endif
  __syncthreads();
  gemm_phase(tile, wpad, wave, m, h, acc);          // key x Wk

  // ---- bias + leaky_relu -> expv[n][k] (lanes with m<4 own cols 0..3) ----
  const float qbv = (m < kK) ? qb[m] : 0.f;
  #pragma unroll
  for (int j = 0; j < 4; ++j) {
    if (m < kK) {
      int n0 = (wave * 4 + j) * 16;
      #pragma unroll
      for (int v = 0; v < 8; ++v) {       // C layout: row = v + 8*h, col = m
        float x = acc[j][v] + qbv;
        x = (x > 0.f) ? x : kSlope * x;
        expv[(n0 + v + 8 * h) * kK + m] = x;
      }
    }
  }
  __syncthreads();

  // ---- softmax over N per head: deterministic tree reductions ----
  const int c  = t & 3;                   // head
  const int r0 = t >> 2;                  // 0..63
  float mx = -3.0e38f;
  #pragma unroll
  for (int i = 0; i < 8; ++i) mx = fmaxf(mx, expv[(r0 + 64 * i) * kK + c]);
  red[t] = mx;
  __syncthreads();
  for (int s = 128; s >= 4; s >>= 1) {
    if (t < s) red[t] = fmaxf(red[t], red[t + s]);
    __syncthreads();
  }
  const float colmax = red[c];
  __syncthreads();

  float lsum = 0.f;
  #pragma unroll
  for (int i = 0; i < 8; ++i) {
    int n = r0 + 64 * i;
    float e = __expf(expv[n * kK + c] - colmax);
    expv[n * kK + c] = e;
    lsum += e;
  }
  red[t] = lsum;
  __syncthreads();
  for (int s = 128; s >= 4; s >>= 1) {
    if (t < s) red[t] += red[t + s];
    __syncthreads();
  }
  float inv[kK];
  #pragma unroll
  for (int k = 0; k < kK; ++k) inv[k] = 1.0f / red[k];

  // w[n] = sum_k attn[n,k]
  for (int n = t; n < kN; n += kThreads) {
    warr[n] = expv[n * kK + 0] * inv[0] + expv[n * kK + 1] * inv[1] +
              expv[n * kK + 2] * inv[2] + expv[n * kK + 3] * inv[3];
  }
  __syncthreads();

  // ---- combine: combined[d] = 0.25 * sum_n w[n]*key[n,d]; key tile is in LDS ----
  const int d = t & 63;
  const int g = t >> 6;                   // 0..3, each sums 128 rows
  float p = 0.f;
  for (int n = g * 128; n < (g + 1) * 128; ++n)
    p = fmaf(warr[n], tile[n * kRS + d], p);
  red[t] = p;                             // red reused: t = g*64 + d
  __syncthreads();
  if (t < kD) {
    float cmb = 0.25f * (red[t] + red[t + 64] + red[t + 128] + red[t + 192]);
    out[(size_t)b * kD + t] = 1.0f / (1.0f + __expf(-cmb));
  }
}

extern "C" void kernel_launch(void* const* d_in, const int* in_sizes, int n_in,
                              void* d_out, int out_size, void* d_ws, size_t ws_size,
                              hipStream_t stream) {
  (void)in_sizes; (void)n_in; (void)out_size; (void)d_ws; (void)ws_size;
  const float* key = (const float*)d_in[0];
  const float* spa = (const float*)d_in[1];
  const float* qg  = (const float*)d_in[2];
  const float* qe  = (const float*)d_in[3];
  const float* av  = (const float*)d_in[4];
  float* out = (float*)d_out;

  const size_t smem =
      (size_t)(kN * kRS + kN * kK + kThreads + kN + kK + 2 * kD * 16) * sizeof(float);
  gpica_attn_kernel<<<kB, kThreads, smem, stream>>>(key, spa, qg, qe, av, out);
}